// HeARBiMambaClassifier_3169685864782
// MI455X (gfx1250) — compile-verified
//
#include <hip/hip_runtime.h>

typedef __attribute__((ext_vector_type(2))) float v2f;
typedef __attribute__((ext_vector_type(8))) float v8f;

#define BM 128
#define BN 128
#define KC 32
#define KCP 36   // padded LDS row stride (floats): conflict-free, keeps 16B align

// ---------------------------------------------------------------------------
// Generic fp32 WMMA GEMM: C[M,N] = op(A)[M,K] @ W[N,K]^T  (+ optional epilogue)
//   A row-major lda, W row-major ldw, C row-major ldc.
//   flipA: A row (b*Lflip + l) read as (b*Lflip + Lflip-1-l)  (sequence flip)
//   epilogue==1: C = softplus(acc + bias[col])
// Block: 256 threads = 8 waves; wave w computes rows [16w,16w+16) x 128 cols.
// Double-buffered LDS tiles filled with GLOBAL_LOAD_ASYNC_TO_LDS_B128
// (ASYNCcnt-tracked), overlapping HBM latency with v_wmma compute.
// ---------------------------------------------------------------------------
__global__ __launch_bounds__(256) void wmma_gemm_f32(
    const float* __restrict__ A, const float* __restrict__ W,
    float* __restrict__ C, const float* __restrict__ bias,
    int M, int N, int K, int lda, int ldw, int ldc,
    int flipA, int Lflip, int epilogue)
{
    __shared__ alignas(16) float As[2][BM * KCP];
    __shared__ alignas(16) float Ws[2][BN * KCP];

    const int tid  = threadIdx.x;
    const int lane = tid & 31;
    const int wv   = tid >> 5;          // wave id 0..7
    const int t16  = lane & 15;
    const int half = lane >> 4;         // 0 or 1
    const int row0 = blockIdx.y * BM;
    const int col0 = blockIdx.x * BN;

    // Issue one K-chunk's tiles as async global->LDS copies (8 instr / wave).
    auto issue_tiles = [&](int k0, int buf) {
        #pragma unroll
        for (int i = 0; i < 4; ++i) {          // A tile: BM x KC
            int idx = tid + i * 256;
            int r   = idx >> 3;                // KC/4 = 8 float4 per row
            int c4  = (idx & 7) << 2;
            int gr  = row0 + r;
            if (flipA) {
                int bb = gr / Lflip;
                int ll = gr - bb * Lflip;
                gr = bb * Lflip + (Lflip - 1 - ll);
            }
            unsigned lds = (unsigned)(size_t)(&As[buf][r * KCP + c4]);
            unsigned long long gp =
                (unsigned long long)(size_t)(&A[(size_t)gr * lda + k0 + c4]);
            asm volatile("global_load_async_to_lds_b128 %0, %1, off"
                         :: "v"(lds), "v"(gp) : "memory");
        }
        #pragma unroll
        for (int i = 0; i < 4; ++i) {          // W tile: BN x KC (clamp rows >= N)
            int idx = tid + i * 256;
            int r   = idx >> 3;
            int c4  = (idx & 7) << 2;
            int gn  = col0 + r;
            if (gn >= N) gn = N - 1;           // garbage cols masked at store
            unsigned lds = (unsigned)(size_t)(&Ws[buf][r * KCP + c4]);
            unsigned long long gp =
                (unsigned long long)(size_t)(&W[(size_t)gn * ldw + k0 + c4]);
            asm volatile("global_load_async_to_lds_b128 %0, %1, off"
                         :: "v"(lds), "v"(gp) : "memory");
        }
    };

    const v8f vzero = {0.f,0.f,0.f,0.f,0.f,0.f,0.f,0.f};
    v8f acc[8];
    #pragma unroll
    for (int nt = 0; nt < 8; ++nt) acc[nt] = vzero;

    const int nch = K / KC;
    issue_tiles(0, 0);

    for (int ch = 0; ch < nch; ++ch) {
        if (ch + 1 < nch) {
            issue_tiles((ch + 1) * KC, (ch + 1) & 1);
            // loads complete in order: <=8 outstanding => chunk ch has landed
            asm volatile("s_wait_asynccnt 0x8" ::: "memory");
        } else {
            asm volatile("s_wait_asynccnt 0x0" ::: "memory");
        }
        __syncthreads();

        const float* Ac = &As[ch & 1][0];
        const float* Wc = &Ws[ch & 1][0];
        const int am = wv * 16;
        #pragma unroll
        for (int kb = 0; kb < KC; kb += 4) {
            // A frag: lane t, elem j -> A[m = t%16, k = kb + 2*(t/16) + j]
            v2f a;
            a.x = Ac[(am + t16) * KCP + kb + 2 * half];
            a.y = Ac[(am + t16) * KCP + kb + 2 * half + 1];
            #pragma unroll
            for (int nt = 0; nt < 8; ++nt) {
                // B frag: lane t, elem j -> W[n = nt*16 + t%16, k = kb + 2*(t/16) + j]
                v2f bfr;
                bfr.x = Wc[(nt * 16 + t16) * KCP + kb + 2 * half];
                bfr.y = Wc[(nt * 16 + t16) * KCP + kb + 2 * half + 1];
                acc[nt] = __builtin_amdgcn_wmma_f32_16x16x4_f32(
                    false, a, false, bfr, (short)0, acc[nt], false, false);
            }
        }
        __syncthreads();   // protect buffer reuse before next chunk's issue
    }

    // ---- store: C frag VGPR v -> m = v + 8*half, n = t16
    const int am = wv * 16;
    #pragma unroll
    for (int nt = 0; nt < 8; ++nt) {
        int col = col0 + nt * 16 + t16;
        if (col >= N) continue;
        float bv = (epilogue == 1) ? bias[col] : 0.f;
        #pragma unroll
        for (int v = 0; v < 8; ++v) {
            int row = row0 + am + v + 8 * half;
            float val = acc[nt][v];
            if (epilogue == 1) {
                val += bv;                                    // dt proj bias
                val = (val > 20.f) ? val : log1pf(expf(val)); // softplus
            }
            C[(size_t)row * ldc + col] = val;
        }
    }
}

// ---------------------------------------------------------------------------
// Depthwise causal conv (k=4) + SiLU: u[b,l,d] = silu(sum_k w[d,k]*xi[b,l-3+k,d]+cb[d])
// xi = first DI columns of xz (row width XW).
// ---------------------------------------------------------------------------
__global__ __launch_bounds__(256) void conv_silu_kernel(
    const float* __restrict__ xz, const float* __restrict__ cw,
    const float* __restrict__ cb, float* __restrict__ u,
    int B, int L, int DI, int XW)
{
    size_t idx = (size_t)blockIdx.x * 256 + threadIdx.x;
    size_t total = (size_t)B * L * DI;
    if (idx >= total) return;
    int d = (int)(idx % DI);
    size_t bl = idx / DI;
    int l = (int)(bl % L);
    int b = (int)(bl / L);

    float acc = cb[d];
    #pragma unroll
    for (int k = 0; k < 4; ++k) {
        int ls = l - 3 + k;
        if (ls >= 0)
            acc += cw[d * 4 + k] * xz[((size_t)b * L + ls) * XW + d];
    }
    u[idx] = acc / (1.f + expf(-acc));   // silu
}

// ---------------------------------------------------------------------------
// Selective scan. One lane per (batch b = blockIdx.y, channel d). Sequential L.
// Writes y = (sum_s h_s*C_s + u*D) * silu(z) IN PLACE over u.
// B_t/C_t (16+16 floats per l) staged in LDS in chunks of 16 timesteps.
// ---------------------------------------------------------------------------
__global__ __launch_bounds__(256) void scan_kernel(
    float* __restrict__ u, const float* __restrict__ dt,
    const float* __restrict__ xdb,   // (B*L, 96): cols 64..79 = B, 80..95 = C
    const float* __restrict__ xz,    // (B*L, 4096): z at col 2048+d
    const float* __restrict__ A_log, const float* __restrict__ Dp,
    int L, int DI)
{
    const int b = blockIdx.y;
    const int d = blockIdx.x * 256 + threadIdx.x;

    float a[16], h[16];
    #pragma unroll
    for (int s = 0; s < 16; ++s) {
        a[s] = -expf(A_log[d * 16 + s]);
        h[s] = 0.f;
    }
    const float Dd = Dp[d];

    __shared__ float sBC[16][32];    // [l-in-chunk][0..15 = B, 16..31 = C]
    const int CL = 16;

    for (int lc0 = 0; lc0 < L; lc0 += CL) {
        __syncthreads();
        for (int i = threadIdx.x; i < CL * 32; i += 256) {
            int li = i >> 5, s = i & 31;
            sBC[li][s] = xdb[((size_t)b * L + lc0 + li) * 96 + 64 + s];
        }
        __syncthreads();
        for (int li = 0; li < CL; ++li) {
            int l = lc0 + li;
            size_t off = ((size_t)b * L + l) * DI + d;
            float dtv = dt[off];
            float uv  = u[off];
            float accy = 0.f;
            #pragma unroll
            for (int s = 0; s < 16; ++s) {
                float dA = __expf(dtv * a[s]);
                h[s] = dA * h[s] + (dtv * sBC[li][s]) * uv;
                accy += h[s] * sBC[li][16 + s];
            }
            float y  = accy + uv * Dd;
            float zv = xz[((size_t)b * L + l) * 4096 + 2048 + d];
            y *= zv / (1.f + __expf(-zv));   // * silu(z)
            u[off] = y;                       // in-place: u becomes yact
        }
    }
}

// ---------------------------------------------------------------------------
// h = out_f[b,l,:] + out_b[b,L-1-l,:]; layernorm over DM; write hn.
// One block per (b,l); 256 threads x 4 channels.
// ---------------------------------------------------------------------------
__global__ __launch_bounds__(256) void ln_kernel(
    const float* __restrict__ outf, const float* __restrict__ outb,
    const float* __restrict__ g, const float* __restrict__ bta,
    float* __restrict__ hn, int L, int DM)
{
    const int b = blockIdx.y;
    const int l = blockIdx.x;
    const int tid = threadIdx.x;
    size_t rf = ((size_t)b * L + l) * DM;
    size_t rb = ((size_t)b * L + (L - 1 - l)) * DM;

    float vals[4];
    float s = 0.f, s2 = 0.f;
    #pragma unroll
    for (int i = 0; i < 4; ++i) {
        int c = tid + i * 256;
        float v = outf[rf + c] + outb[rb + c];
        vals[i] = v;
        s += v; s2 += v * v;
    }
    __shared__ float rs[256], rs2[256], bc[2];
    rs[tid] = s; rs2[tid] = s2;
    __syncthreads();
    for (int st = 128; st > 0; st >>= 1) {
        if (tid < st) { rs[tid] += rs[tid + st]; rs2[tid] += rs2[tid + st]; }
        __syncthreads();
    }
    if (tid == 0) {
        float mu  = rs[0] / (float)DM;
        float var = rs2[0] / (float)DM - mu * mu;
        bc[0] = mu;
        bc[1] = rsqrtf(var + 1e-5f);
    }
    __syncthreads();
    float mu = bc[0], rsig = bc[1];
    #pragma unroll
    for (int i = 0; i < 4; ++i) {
        int c = tid + i * 256;
        hn[rf + c] = (vals[i] - mu) * rsig * g[c] + bta[c];
    }
}

// combined[b,c] = mean_l hn + max_l hn
__global__ __launch_bounds__(256) void combine_kernel(
    const float* __restrict__ hn, float* __restrict__ comb, int L, int DM)
{
    const int b = blockIdx.y;
    const int c = blockIdx.x * 256 + threadIdx.x;
    float s = 0.f, m = -3.4e38f;
    for (int l = 0; l < L; ++l) {
        float v = hn[((size_t)b * L + l) * DM + c];
        s += v;
        m = fmaxf(m, v);
    }
    comb[b * DM + c] = s / (float)L + m;
}

// fc1 -> bn(scale) -> exact gelu -> fc2. One block per batch, thread per hidden j.
__global__ __launch_bounds__(512) void head_kernel(
    const float* __restrict__ comb, const float* __restrict__ fc1w,
    const float* __restrict__ fc1b, const float* __restrict__ bng,
    const float* __restrict__ bnb, const float* __restrict__ fc2w,
    const float* __restrict__ fc2b, float* __restrict__ out, int DM)
{
    const int b = blockIdx.x;
    const int j = threadIdx.x;
    const float* cb = comb + b * DM;
    float acc = 0.f;
    for (int c = 0; c < DM; ++c) acc += cb[c] * fc1w[j * DM + c];
    acc += fc1b[j];
    float z  = acc * rsqrtf(1.f + 1e-5f) * bng[j] + bnb[j];
    float ge = 0.5f * z * (1.f + erff(z * 0.70710678118654752f));
    __shared__ float red[512];
    red[j] = ge * fc2w[j];
    __syncthreads();
    for (int st = 256; st > 0; st >>= 1) {
        if (j < st) red[j] += red[j + st];
        __syncthreads();
    }
    if (j == 0) out[b] = red[0] + fc2b[0];
}

// ---------------------------------------------------------------------------
extern "C" void kernel_launch(void* const* d_in, const int* in_sizes, int n_in,
                              void* d_out, int out_size, void* d_ws, size_t ws_size,
                              hipStream_t stream)
{
    (void)in_sizes; (void)n_in; (void)out_size; (void)ws_size;
    const int Bsz = 4, L = 1024, DM = 1024, DI = 2048, XW = 4096;
    const int M = Bsz * L;                // 4096 rows

    const float* x = (const float*)d_in[0];
    const float* W_in[2]   = {(const float*)d_in[1],  (const float*)d_in[10]};
    const float* conv_w[2] = {(const float*)d_in[2],  (const float*)d_in[11]};
    const float* conv_b[2] = {(const float*)d_in[3],  (const float*)d_in[12]};
    const float* W_xp[2]   = {(const float*)d_in[4],  (const float*)d_in[13]};
    const float* W_dt[2]   = {(const float*)d_in[5],  (const float*)d_in[14]};
    const float* b_dt[2]   = {(const float*)d_in[6],  (const float*)d_in[15]};
    const float* A_log[2]  = {(const float*)d_in[7],  (const float*)d_in[16]};
    const float* Dp[2]     = {(const float*)d_in[8],  (const float*)d_in[17]};
    const float* W_out[2]  = {(const float*)d_in[9],  (const float*)d_in[18]};
    const float* ln_g = (const float*)d_in[19];
    const float* ln_b = (const float*)d_in[20];
    const float* fc1w = (const float*)d_in[21];
    const float* fc1b = (const float*)d_in[22];
    const float* bng  = (const float*)d_in[23];
    const float* bnb  = (const float*)d_in[24];
    const float* fc2w = (const float*)d_in[25];
    const float* fc2b = (const float*)d_in[26];

    float* ws = (float*)d_ws;
    size_t off = 0;
    auto alloc = [&](size_t n) { float* p = ws + off; off += n; return p; };
    float* xz[2]  = {alloc((size_t)M * XW), alloc((size_t)M * XW)};   // 64MB ea
    float* u[2]   = {alloc((size_t)M * DI), alloc((size_t)M * DI)};   // 32MB ea (becomes yact)
    float* xdb[2] = {alloc((size_t)M * 96), alloc((size_t)M * 96)};
    float* dtb[2] = {alloc((size_t)M * DI), alloc((size_t)M * DI)};
    float* outd[2]= {alloc((size_t)M * DM), alloc((size_t)M * DM)};
    float* hn   = xz[0];                     // dead after scans -> reuse
    float* comb = xz[1];

    for (int dir = 0; dir < 2; ++dir) {
        // xz = (flip?)x @ W_in^T : [4096,4096] K=1024
        wmma_gemm_f32<<<dim3(32, 32), 256, 0, stream>>>(
            x, W_in[dir], xz[dir], nullptr,
            M, 4096, 1024, DM, DM, XW, dir, L, 0);
        // depthwise conv + silu -> u
        conv_silu_kernel<<<(unsigned)(((size_t)M * DI) / 256), 256, 0, stream>>>(
            xz[dir], conv_w[dir], conv_b[dir], u[dir], Bsz, L, DI, XW);
        // xdb = u @ W_xproj^T : [4096,96] K=2048
        wmma_gemm_f32<<<dim3(1, 32), 256, 0, stream>>>(
            u[dir], W_xp[dir], xdb[dir], nullptr,
            M, 96, DI, DI, DI, 96, 0, L, 0);
        // dt = softplus(xdb[:, :64] @ W_dt^T + b_dt) : [4096,2048] K=64
        wmma_gemm_f32<<<dim3(16, 32), 256, 0, stream>>>(
            xdb[dir], W_dt[dir], dtb[dir], b_dt[dir],
            M, DI, 64, 96, 64, DI, 0, L, 1);
        // selective scan (in-place u -> yact)
        scan_kernel<<<dim3(DI / 256, Bsz), 256, 0, stream>>>(
            u[dir], dtb[dir], xdb[dir], xz[dir], A_log[dir], Dp[dir], L, DI);
        // out = yact @ W_out^T : [4096,1024] K=2048
        wmma_gemm_f32<<<dim3(8, 32), 256, 0, stream>>>(
            u[dir], W_out[dir], outd[dir], nullptr,
            M, DM, DI, DI, DI, DM, 0, L, 0);
    }

    ln_kernel<<<dim3(L, Bsz), 256, 0, stream>>>(outd[0], outd[1], ln_g, ln_b, hn, L, DM);
    combine_kernel<<<dim3(DM / 256, Bsz), 256, 0, stream>>>(hn, comb, L, DM);
    head_kernel<<<Bsz, 512, 0, stream>>>(comb, fc1w, fc1b, bng, bnb, fc2w, fc2b,
                                         (float*)d_out, DM);
}